// CTCLayer_85976655331605
// MI455X (gfx1250) — compile-verified
//
#include <hip/hip_runtime.h>
#include <hip/hip_bf16.h>
#include <math.h>

typedef float v2f __attribute__((ext_vector_type(2)));
typedef float v8f __attribute__((ext_vector_type(8)));

#define CTC_EPS  1e-7f
#define CTC_NEG  (-1.0e30f)
#define CTC_B    512
#define CTC_T    512
#define CTC_C    128
#define CTC_U    64
#define CTC_S    129   // 2U+1

// ---------------------------------------------------------------------------
// Kernel A: rowsum[r] = sum_c (y_pred[r*128+c] + EPS), r in [0, B*T)
// One wave32 per 16 rows; reduction done on the matrix pipe with
// V_WMMA_F32_16X16X4_F32 against an all-ones B matrix (exact f32 MACs).
// A-matrix 16x4 f32 layout: lanes 0-15 hold (K0,K1) of row M=lane,
// lanes 16-31 hold (K2,K3) of row M=lane-16  -> each lane loads a float2.
// D layout: VGPR r, lanes 0-15 => M=r; lanes 16-31 => M=8+r (all N equal).
// ---------------------------------------------------------------------------
__global__ __launch_bounds__(256) void ctc_rowsum_wmma(
    const float* __restrict__ p, float* __restrict__ rowsum, int nrows) {
  int wave = (int)((blockIdx.x * blockDim.x + threadIdx.x) >> 5);
  int lane = (int)(threadIdx.x & 31);
  int row0 = wave * 16;
  if (row0 >= nrows) return;  // wave-uniform: EXEC stays all-ones for WMMA

  int m       = lane & 15;          // row within the 16-row tile
  int colbase = (lane >> 4) * 2;    // 0 for lanes 0-15, 2 for lanes 16-31
  const float* rp = p + (size_t)(row0 + m) * CTC_C + colbase;

  v2f bones; bones.x = 1.0f; bones.y = 1.0f;
  v8f acc = {};
#pragma unroll
  for (int k = 0; k < 32; ++k) {           // K=4 per WMMA, 32 chunks => C=128
    v2f a = *(const v2f*)(rp + k * 4);
    a.x += CTC_EPS;
    a.y += CTC_EPS;
    acc = __builtin_amdgcn_wmma_f32_16x16x4_f32(
        /*neg_a=*/false, a, /*neg_b=*/false, bones,
        /*c_mod=*/(short)0, acc, /*reuse_a=*/false, /*reuse_b=*/false);
  }

  if (m == 0) {                            // lanes 0 (rows 0-7) and 16 (rows 8-15)
    int mbase = (lane >> 4) * 8;
#pragma unroll
    for (int r = 0; r < 8; ++r) rowsum[row0 + mbase + r] = acc[r];
  }
}

// ---------------------------------------------------------------------------
// Kernel B: CTC forward DP. One 160-thread block (5 wave32) per batch sample.
// Threads 0..128 own extended-label states. Emissions per (b,t):
//   emit[s] = log(p[b,t,class(s)] + eps) - log(rowsum[b,t])
// (log_softmax(log x) == log x - log(sum x); no exp needed for emissions.)
// ---------------------------------------------------------------------------
__global__ __launch_bounds__(160) void ctc_forward_dp(
    const float* __restrict__ p, const int* __restrict__ labels,
    const float* __restrict__ rowsum, float* __restrict__ out) {
  __shared__ int   lab[CTC_U];
  __shared__ float ll[CTC_U + 1];    // [0..63]=label logs, [64]=blank log
  __shared__ float lz;               // log(rowsum) for current t
  __shared__ float alpha[2][CTC_S];

  const int b   = (int)blockIdx.x;
  const int tid = (int)threadIdx.x;
  const float* pb = p + (size_t)b * CTC_T * CTC_C;
  const float* rs = rowsum + (size_t)b * CTC_T;

  if (tid < CTC_U) lab[tid] = labels[b * CTC_U + tid];
  __syncthreads();

  const bool isState = (tid < CTC_S);
  bool skip = false;                 // allow s-2 -> s transition
  int  labIdx = -1;                  // -1 => blank state
  if (isState && (tid & 1)) {
    labIdx = (tid - 1) >> 1;
    if (tid >= 3) skip = (lab[labIdx] != lab[(tid - 3) >> 1]);
  }

  int cur = 0;
  for (int t = 0; t < CTC_T; ++t) {
    const float* rowp = pb + (size_t)t * CTC_C;

    // --- emission phase: <=66 distinct logs per (b,t) ---
    if (tid < CTC_U) {
      ll[tid] = __logf(rowp[lab[tid]] + CTC_EPS);
    } else if (tid == CTC_U) {
      ll[CTC_U] = __logf(rowp[CTC_C - 1] + CTC_EPS);  // blank = C-1
    } else if (tid == CTC_U + 1) {
      lz = __logf(rs[t]);
    }
    // prefetch next row while this step computes (global_prefetch_b8)
    if (t + 1 < CTC_T && tid >= 128 && tid < 132) {
      __builtin_prefetch(rowp + CTC_C + (tid - 128) * 32, 0, 3);
    }
    __syncthreads();

    // --- state-update phase ---
    float an = CTC_NEG;
    if (isState) {
      float e = ((tid & 1) ? ll[labIdx] : ll[CTC_U]) - lz;
      if (t == 0) {
        an = (tid < 2) ? e : CTC_NEG;
      } else {
        float a0 = alpha[cur][tid];
        float a1 = (tid >= 1) ? alpha[cur][tid - 1] : CTC_NEG;
        float a2 = skip ? alpha[cur][tid - 2] : CTC_NEG;
        float mx = fmaxf(a0, fmaxf(a1, a2));
        an = mx + __logf(__expf(a0 - mx) + __expf(a1 - mx) + __expf(a2 - mx)) + e;
      }
      alpha[cur ^ 1][tid] = an;
    }
    cur ^= 1;
    __syncthreads();   // protects alpha[cur] reads and ll rewrites next iter
  }

  if (tid == 0) {
    float aA = alpha[cur][CTC_S - 1];
    float aB = alpha[cur][CTC_S - 2];
    float mx = fmaxf(aA, aB);
    out[b] = -(mx + __logf(__expf(aA - mx) + __expf(aB - mx)));
  }
}

// ---------------------------------------------------------------------------
extern "C" void kernel_launch(void* const* d_in, const int* in_sizes, int n_in,
                              void* d_out, int out_size, void* d_ws, size_t ws_size,
                              hipStream_t stream) {
  (void)in_sizes; (void)n_in; (void)out_size; (void)ws_size;
  const int*   y_true = (const int*)d_in[0];    // [B, U] int32
  const float* y_pred = (const float*)d_in[1];  // [B, T, C] float32
  float*       out    = (float*)d_out;          // [B, 1] float32
  float*       rowsum = (float*)d_ws;           // B*T floats = 1 MB scratch

  const int nrows = CTC_B * CTC_T;              // 262144 rows of C=128
  // 16 rows per wave, 8 waves (256 threads) per block -> 2048 blocks
  ctc_rowsum_wmma<<<nrows / (16 * 8), 256, 0, stream>>>(y_pred, rowsum, nrows);
  ctc_forward_dp<<<CTC_B, 160, 0, stream>>>(y_pred, y_true, rowsum, out);
}